// mTransformerConv_61237643706852
// MI455X (gfx1250) — compile-verified
//
#include <hip/hip_runtime.h>
#include <math.h>

#define NN 100000
#define NE 1600000
#define NG 64
#define EPS_SM 1e-16f
#define EPS_GN 1e-5f

typedef __attribute__((ext_vector_type(2))) float v2f;
typedef __attribute__((ext_vector_type(8))) float v8f;

// ---- order-preserving float<->uint key (for atomic segment-max) ----
__device__ __forceinline__ unsigned f2key(float f) {
  unsigned u = __float_as_uint(f);
  return (u & 0x80000000u) ? ~u : (u | 0x80000000u);
}
__device__ __forceinline__ float key2f(unsigned k) {
  unsigned u = (k & 0x80000000u) ? (k & 0x7fffffffu) : ~k;
  return __uint_as_float(u);
}

__device__ __forceinline__ int imin(int a, int b) { return a < b ? a : b; }

// ---------------- zero fill ----------------
__global__ void zero_u32_kernel(unsigned* p, size_t n) {
  size_t i = (size_t)blockIdx.x * blockDim.x + threadIdx.x;
  if (i < n) p[i] = 0u;
}

// ---------------- WMMA f32 GEMM: Y[nrows,F](stride LDY) = X[nrows,K](stride LDX) @ W + bias
// Requirements: LDX >= Kpad(=round4(K)); X columns K..Kpad-1 must be zero (writer pads).
// W staged transposed + zero-padded in LDS -> unconditional ds_load_b64 B-fragments.
// One wave: 4 consecutive 16-row M-tiles x one 16-col N-tile; EXEC all-1s everywhere.
template <int K, int LDX, int F, int LDY>
__global__ __launch_bounds__(128)
void linear_wmma_kernel(const float* __restrict__ X, const float* __restrict__ Wm,
                        const float* __restrict__ bias, float* __restrict__ Y,
                        int nrows, int nwaves) {
  constexpr int KP  = (K + 3) & ~3;
  constexpr int NTN = (F + 15) / 16;
  constexpr int FP  = NTN * 16;
  static_assert(LDX >= KP, "row stride must cover padded K");
  __shared__ float sWt[FP * KP];   // sWt[c*KP + k], zero padded
  __shared__ float sB[FP];
  {
    int t = threadIdx.y * 32 + threadIdx.x;
    for (int i = t; i < FP * KP; i += 128) {
      int c = i / KP, kk = i % KP;
      sWt[i] = (c < F && kk < K) ? Wm[kk * F + c] : 0.0f;
    }
    for (int i = t; i < FP; i += 128) sB[i] = (i < F) ? bias[i] : 0.0f;
  }
  __syncthreads();

  int lane = threadIdx.x;                    // 0..31
  int wid  = blockIdx.x * 4 + threadIdx.y;
  if (wid >= nwaves) return;                 // uniform per wave
  int mg = wid / NTN, tn = wid % NTN;

  int row0  = mg * 64;
  int colB  = tn * 16 + (lane & 15);
  int khalf = (lane >> 4) * 2;               // 0 or 2

  const float* pb = &sWt[colB * KP + khalf];
  const float* pa0 = X + (size_t)imin(row0 +  0 + (lane & 15), nrows - 1) * LDX + khalf;
  const float* pa1 = X + (size_t)imin(row0 + 16 + (lane & 15), nrows - 1) * LDX + khalf;
  const float* pa2 = X + (size_t)imin(row0 + 32 + (lane & 15), nrows - 1) * LDX + khalf;
  const float* pa3 = X + (size_t)imin(row0 + 48 + (lane & 15), nrows - 1) * LDX + khalf;

  v8f acc0 = {}, acc1 = {}, acc2 = {}, acc3 = {};
#pragma unroll
  for (int k0 = 0; k0 < KP; k0 += 4) {
    v2f b  = *(const v2f*)(pb  + k0);
    v2f a0 = *(const v2f*)(pa0 + k0);
    v2f a1 = *(const v2f*)(pa1 + k0);
    v2f a2 = *(const v2f*)(pa2 + k0);
    v2f a3 = *(const v2f*)(pa3 + k0);
    acc0 = __builtin_amdgcn_wmma_f32_16x16x4_f32(false, a0, false, b, (short)0, acc0, false, false);
    acc1 = __builtin_amdgcn_wmma_f32_16x16x4_f32(false, a1, false, b, (short)0, acc1, false, false);
    acc2 = __builtin_amdgcn_wmma_f32_16x16x4_f32(false, a2, false, b, (short)0, acc2, false, false);
    acc3 = __builtin_amdgcn_wmma_f32_16x16x4_f32(false, a3, false, b, (short)0, acc3, false, false);
  }
  // D layout: VGPR i -> (M = i + 8*(lane>>4), N = lane&15)
  int scol = tn * 16 + (lane & 15);
  if (scol < F) {
    float bb = sB[scol];
#pragma unroll
    for (int mt = 0; mt < 4; ++mt) {
      v8f acc = (mt == 0) ? acc0 : (mt == 1) ? acc1 : (mt == 2) ? acc2 : acc3;
      int srow0 = row0 + mt * 16 + ((lane >> 4) << 3);
#pragma unroll
      for (int i = 0; i < 8; ++i) {
        int rr = srow0 + i;
        if (rr < nrows) Y[(size_t)rr * LDY + scol] = acc[i] + bb;
      }
    }
  }
}

// ---------------- edge pass 1: alpha + segment max (sortable-key atomic max) ----------------
template <int H, int C, int LD>
__global__ __launch_bounds__(256)
void edge_alpha_kernel(const float* __restrict__ q, const float* __restrict__ k,
                       const float* __restrict__ ea, const float* __restrict__ We,
                       const int* __restrict__ src, const int* __restrict__ dst,
                       float* __restrict__ alpha, unsigned* __restrict__ amax,
                       float scale) {
  constexpr int F = H * C;
  __shared__ float sWe[8 * F];
  for (int i = threadIdx.x; i < 8 * F; i += 256) sWe[i] = We[i];
  __syncthreads();
  int e = blockIdx.x * 256 + threadIdx.x;
  if (e >= NE) return;
  int s = src[e], d = dst[e];
  float a8[8];
#pragma unroll
  for (int j = 0; j < 8; ++j) a8[j] = ea[(size_t)e * 8 + j];
  const float* qr = q + (size_t)d * LD;
  const float* kr = k + (size_t)s * LD;
  for (int h = 0; h < H; ++h) {
    float al = 0.0f;
#pragma unroll
    for (int c = 0; c < C; ++c) {
      int f = h * C + c;
      float ev = 0.0f;
#pragma unroll
      for (int j = 0; j < 8; ++j) ev += a8[j] * sWe[j * F + f];
      al += qr[f] * (kr[f] + ev);
    }
    al *= scale;
    alpha[(size_t)e * H + h] = al;
    atomicMax(&amax[(size_t)d * H + h], f2key(al));
  }
}

// ---------------- edge pass 2: exp + fused unnormalized accumulation ----------------
template <int H, int C, int LD>
__global__ __launch_bounds__(256)
void edge_accum_kernel(const float* __restrict__ v, const float* __restrict__ ea,
                       const float* __restrict__ We,
                       const int* __restrict__ src, const int* __restrict__ dst,
                       const float* __restrict__ alpha, const unsigned* __restrict__ amax,
                       float* __restrict__ den, float* __restrict__ agg) {
  constexpr int F = H * C;
  __shared__ float sWe[8 * F];
  for (int i = threadIdx.x; i < 8 * F; i += 256) sWe[i] = We[i];
  __syncthreads();
  int e = blockIdx.x * 256 + threadIdx.x;
  if (e >= NE) return;
  int s = src[e], d = dst[e];
  float a8[8];
#pragma unroll
  for (int j = 0; j < 8; ++j) a8[j] = ea[(size_t)e * 8 + j];
  const float* vr = v + (size_t)s * LD;
  for (int h = 0; h < H; ++h) {
    float m  = key2f(amax[(size_t)d * H + h]);
    float ex = __expf(alpha[(size_t)e * H + h] - m);
    atomicAdd(&den[(size_t)d * H + h], ex);
#pragma unroll
    for (int c = 0; c < C; ++c) {
      int f = h * C + c;
      float ev = 0.0f;
#pragma unroll
      for (int j = 0; j < 8; ++j) ev += a8[j] * sWe[j * F + f];
      atomicAdd(&agg[(size_t)d * LD + f], ex * (vr[f] + ev));
    }
  }
}

// ---------------- per-node finish: normalize, concat/mean, beta gate ----------------
template <int H, int C, int CONCAT, int LD, int LDO>
__global__ __launch_bounds__(256)
void node_finish_kernel(const float* __restrict__ agg, const float* __restrict__ den,
                        const float* __restrict__ r, const float* __restrict__ Wb,
                        float* __restrict__ out, int do_sigmoid) {
  constexpr int F = H * C;
  constexpr int S = CONCAT ? F : C;
  int n = blockIdx.x * 256 + threadIdx.x;
  if (n >= NN) return;
  const float* ag = agg + (size_t)n * LD;
  const float* dn = den + (size_t)n * H;
  const float* rr = r + (size_t)n * LD;
  constexpr float invH = 1.0f / (float)H;
  float bl = 0.0f;
#pragma unroll
  for (int sIdx = 0; sIdx < S; ++sIdx) {
    float o;
    if (CONCAT) {
      o = ag[sIdx] / (dn[sIdx / C] + EPS_SM);
    } else {
      float acc = 0.0f;
#pragma unroll
      for (int h = 0; h < H; ++h) acc += ag[h * C + sIdx] / (dn[h] + EPS_SM);
      o = acc * invH;
    }
    float rv = rr[sIdx];
    bl += o * Wb[sIdx] + rv * Wb[S + sIdx] + (o - rv) * Wb[2 * S + sIdx];
  }
  float beta = 1.0f / (1.0f + __expf(-bl));
#pragma unroll
  for (int sIdx = 0; sIdx < S; ++sIdx) {
    float o;
    if (CONCAT) {
      o = ag[sIdx] / (dn[sIdx / C] + EPS_SM);
    } else {
      float acc = 0.0f;
#pragma unroll
      for (int h = 0; h < H; ++h) acc += ag[h * C + sIdx] / (dn[h] + EPS_SM);
      o = acc * invH;
    }
    float y = beta * rr[sIdx] + (1.0f - beta) * o;
    if (do_sigmoid) y = 1.0f / (1.0f + __expf(-y));
    out[(size_t)n * LDO + sIdx] = y;
  }
}

// ---------------- GraphNorm passes ----------------
template <int F, int LDI>
__global__ __launch_bounds__(256)
void gn_accum_kernel(const float* __restrict__ h, const int* __restrict__ bat,
                     float* __restrict__ gsum, float* __restrict__ gcnt) {
  int n = blockIdx.x * 256 + threadIdx.x;
  if (n >= NN) return;
  int g = bat[n];
  atomicAdd(&gcnt[g], 1.0f);
#pragma unroll
  for (int f = 0; f < F; ++f) atomicAdd(&gsum[(size_t)g * F + f], h[(size_t)n * LDI + f]);
}

template <int F, int LDI>
__global__ __launch_bounds__(256)
void gn_center_kernel(const float* __restrict__ h, const int* __restrict__ bat,
                      const float* __restrict__ gsum, const float* __restrict__ gcnt,
                      const float* __restrict__ ms, float* __restrict__ cen,
                      float* __restrict__ gvar) {
  int n = blockIdx.x * 256 + threadIdx.x;
  if (n >= NN) return;
  int g = bat[n];
  float ic = 1.0f / gcnt[g];
#pragma unroll
  for (int f = 0; f < F; ++f) {
    float c = h[(size_t)n * LDI + f] - ms[f] * (gsum[(size_t)g * F + f] * ic);
    cen[(size_t)n * LDI + f] = c;
    atomicAdd(&gvar[(size_t)g * F + f], c * c);
  }
}

// writes zeros into pad columns F..LDO-1 so the next GEMM can read Kpad unconditionally
template <int F, int LDI, int LDO>
__global__ __launch_bounds__(256)
void gn_final_kernel(const float* __restrict__ cen, const int* __restrict__ bat,
                     const float* __restrict__ gvar, const float* __restrict__ gcnt,
                     const float* __restrict__ w, const float* __restrict__ b,
                     float* __restrict__ out) {
  int n = blockIdx.x * 256 + threadIdx.x;
  if (n >= NN) return;
  int g = bat[n];
  float ic = 1.0f / gcnt[g];
#pragma unroll
  for (int f = 0; f < LDO; ++f) {
    float y = 0.0f;
    if (f < F) {
      y = w[f] * cen[(size_t)n * LDI + f] *
          rsqrtf(gvar[(size_t)g * F + f] * ic + EPS_GN) + b[f];
      y = fmaxf(y, 0.0f);
    }
    out[(size_t)n * LDO + f] = y;
  }
}

// ---------------- host-side orchestration ----------------
static inline void zero_buf(hipStream_t st, void* p, size_t n_u32) {
  int blk = (int)((n_u32 + 255) / 256);
  zero_u32_kernel<<<blk, 256, 0, st>>>((unsigned*)p, n_u32);
}

template <int K, int LDX, int F, int LDY>
static inline void run_linear(hipStream_t st, const float* X, const float* Wm,
                              const float* bias, float* Y, int nrows) {
  constexpr int NTN = (F + 15) / 16;
  int mgroups = (nrows + 63) / 64;
  int nwaves = mgroups * NTN;
  int blocks = (nwaves + 3) / 4;
  linear_wmma_kernel<K, LDX, F, LDY><<<dim3(blocks), dim3(32, 4), 0, st>>>(
      X, Wm, bias, Y, nrows, nwaves);
}

struct Bufs {
  float *q, *k, *v, *r, *agg, *hA, *hB, *alpha, *den;
  unsigned* amax;
  float *gsum, *gvar, *gcnt;
};

template <int DIN, int LDX, int H, int C, int CONCAT, int LD, int LDO>
static void run_tconv(hipStream_t st, const float* xin, void* const* d_in, int base,
                      const int* src, const int* dst, const float* ea, Bufs& B,
                      float* out, int do_sigmoid) {
  const float* Wq = (const float*)d_in[base + 0];
  const float* bq = (const float*)d_in[base + 1];
  const float* Wk = (const float*)d_in[base + 2];
  const float* bk = (const float*)d_in[base + 3];
  const float* Wv = (const float*)d_in[base + 4];
  const float* bv = (const float*)d_in[base + 5];
  const float* We = (const float*)d_in[base + 6];
  const float* Ws = (const float*)d_in[base + 7];
  const float* bs = (const float*)d_in[base + 8];
  const float* Wb = (const float*)d_in[base + 9];
  constexpr int F = H * C;
  constexpr int S = CONCAT ? F : C;
  static_assert(S == F || !CONCAT, "");
  run_linear<DIN, LDX, F, LD>(st, xin, Wq, bq, B.q, NN);
  run_linear<DIN, LDX, F, LD>(st, xin, Wk, bk, B.k, NN);
  run_linear<DIN, LDX, F, LD>(st, xin, Wv, bv, B.v, NN);
  run_linear<DIN, LDX, S, LD>(st, xin, Ws, bs, B.r, NN);
  zero_buf(st, B.amax, (size_t)NN * H);
  zero_buf(st, B.den,  (size_t)NN * H);
  zero_buf(st, B.agg,  (size_t)NN * LD);
  float scale = 1.0f / sqrtf((float)C);
  int eb = (NE + 255) / 256;
  edge_alpha_kernel<H, C, LD><<<eb, 256, 0, st>>>(B.q, B.k, ea, We, src, dst, B.alpha,
                                                  B.amax, scale);
  edge_accum_kernel<H, C, LD><<<eb, 256, 0, st>>>(B.v, ea, We, src, dst, B.alpha,
                                                  B.amax, B.den, B.agg);
  int nb = (NN + 255) / 256;
  node_finish_kernel<H, C, CONCAT, LD, LDO><<<nb, 256, 0, st>>>(B.agg, B.den, B.r, Wb,
                                                                out, do_sigmoid);
}

template <int F, int LDI, int LDO>
static void run_gnorm(hipStream_t st, const float* hin, const int* bat,
                      const float* w, const float* b, const float* ms,
                      Bufs& B, float* out) {
  zero_buf(st, B.gsum, (size_t)NG * F);
  zero_buf(st, B.gvar, (size_t)NG * F);
  zero_buf(st, B.gcnt, (size_t)NG);
  int nb = (NN + 255) / 256;
  gn_accum_kernel<F, LDI><<<nb, 256, 0, st>>>(hin, bat, B.gsum, B.gcnt);
  gn_center_kernel<F, LDI><<<nb, 256, 0, st>>>(hin, bat, B.gsum, B.gcnt, ms, B.q, B.gvar);
  gn_final_kernel<F, LDI, LDO><<<nb, 256, 0, st>>>(B.q, bat, B.gvar, B.gcnt, w, b, out);
}

extern "C" void kernel_launch(void* const* d_in, const int* in_sizes, int n_in,
                              void* d_out, int out_size, void* d_ws, size_t ws_size,
                              hipStream_t stream) {
  const float* x   = (const float*)d_in[0];
  const int*   ei  = (const int*)d_in[1];
  const float* ea  = (const float*)d_in[2];
  const int*   bat = (const int*)d_in[3];
  const int* src = ei;        // edge_index[0,:]
  const int* dst = ei + NE;   // edge_index[1,:]

  // workspace layout (floats); feature buffers use padded stride 52
  size_t nf = (size_t)NN * 52;
  float* base = (float*)d_ws;
  Bufs B;
  B.q     = base;
  B.k     = B.q + nf;
  B.v     = B.k + nf;
  B.r     = B.v + nf;
  B.agg   = B.r + nf;
  B.hA    = B.agg + nf;
  B.hB    = B.hA + nf;
  B.alpha = B.hB + nf;                    // E*10
  B.den   = B.alpha + (size_t)NE * 10;    // N*10
  B.amax  = (unsigned*)(B.den + (size_t)NN * 10);  // N*10
  B.gsum  = (float*)(B.amax + (size_t)NN * 10);    // G*50
  B.gvar  = B.gsum + (size_t)NG * 50;
  B.gcnt  = B.gvar + (size_t)NG * 50;              // G

  // layer 1: din=32 (x stride 32), H=10, C=5, concat -> hA (F=50, stride 52)
  run_tconv<32, 32, 10, 5, 1, 52, 52>(stream, x, d_in, 4, src, dst, ea, B, B.hA, 0);
  // gn1 (F=50, in stride 52) -> hB stride 52 with zeroed pad cols (Kpad=52 for next GEMM)
  run_gnorm<50, 52, 52>(stream, B.hA, bat, (const float*)d_in[34],
                        (const float*)d_in[35], (const float*)d_in[36], B, B.hB);
  // layer 2: din=50 (hB stride 52), H=10, C=2, concat -> hA (F=20, stride 20)
  run_tconv<50, 52, 10, 2, 1, 20, 20>(stream, B.hB, d_in, 14, src, dst, ea, B, B.hA, 0);
  // gn2 (F=20) -> hB stride 20 (20 is already a multiple of 4)
  run_gnorm<20, 20, 20>(stream, B.hA, bat, (const float*)d_in[37],
                        (const float*)d_in[38], (const float*)d_in[39], B, B.hB);
  // layer 3: din=20 (hB stride 20), H=1, C=1, mean -> d_out (stride 1) + sigmoid
  run_tconv<20, 20, 1, 1, 0, 1, 1>(stream, B.hB, d_in, 24, src, dst, ea, B,
                                   (float*)d_out, 1);
}